// Net_47115791237865
// MI455X (gfx1250) — compile-verified
//
#include <hip/hip_runtime.h>
#include <math.h>

typedef __attribute__((ext_vector_type(2))) float v2f;
typedef __attribute__((ext_vector_type(8))) float v8f;

#define N_NODES 12288
#define N_EDGES 393216
#define DIM     256
#define KCLUST  16

// ---------------------------------------------------------------------------
// fp32 WMMA GEMM: C[M,Nc] = A[M,K] @ B[K,Nc] (+ bias broadcast over rows).
// One wave computes a 16 x (16*NT) strip via V_WMMA_F32_16X16X4_F32.
// A-matrix VGPR layout (ISA 7.12.2, 32-bit A 16x4): lanes 0-15 hold K=0,1
// (M=lane); lanes 16-31 hold K=2,3 (M=lane-16).  The per-half K offset
// (0 or 2) is folded into the A/B base pointers BEFORE the loop so the
// steady state is: one b64 A load, NT pairs of B loads, NT WMMAs, two
// pointer adds - no cndmask / mul in the hot loop.
// C/D: VGPR r -> row r (lanes 0-15) / row r+8 (lanes 16-31), col = lane&15.
// Grids are sized exactly (no guards) so EXEC is all-ones at every WMMA.
// ---------------------------------------------------------------------------
template <int NT>
__global__ __launch_bounds__(256) void gemm_f32_wmma(
    const float* __restrict__ A, const float* __restrict__ B,
    const float* __restrict__ bias, float* __restrict__ C,
    int M, int K, int Nc)
{
    const int lane   = threadIdx.x & 31;
    const int wave   = threadIdx.x >> 5;
    const int tilesN = Nc / (16 * NT);
    const int tile   = blockIdx.x * 8 + wave;
    const int m0     = (tile / tilesN) * 16;
    const int n0     = (tile % tilesN) * (16 * NT);
    const int ln     = lane & 15;
    const int koff   = (lane >> 4) << 1;       // 0 (lanes 0-15) or 2 (16-31)

    v8f c[NT];
    #pragma unroll
    for (int t = 0; t < NT; ++t) {
        const float bi = bias ? bias[n0 + 16 * t + ln] : 0.0f;
        #pragma unroll
        for (int r = 0; r < 8; ++r) c[t][r] = bi;
    }

    const float* ap = A + (size_t)(m0 + ln) * K + koff;       // 8B-aligned
    const float* bp = B + (size_t)koff * Nc + n0 + ln;
    const size_t bstep = (size_t)4 * Nc;

    for (int k0 = 0; k0 < K; k0 += 4) {
        const v2f a = *(const v2f*)ap;         // global_load_b64
        #pragma unroll
        for (int t = 0; t < NT; ++t) {
            v2f b;
            b.x = bp[16 * t];
            b.y = bp[16 * t + Nc];
            c[t] = __builtin_amdgcn_wmma_f32_16x16x4_f32(
                       /*neg_a=*/false, a, /*neg_b=*/false, b,
                       /*c_mod=*/(short)0, c[t], /*reuse_a=*/false, /*reuse_b=*/false);
        }
        ap += 4;
        bp += bstep;
    }

    const int mrow = m0 + ((lane >> 4) << 3);  // +8 rows for upper half-wave
    #pragma unroll
    for (int t = 0; t < NT; ++t) {
        float* crow = C + (size_t)mrow * Nc + n0 + 16 * t + ln;
        #pragma unroll
        for (int r = 0; r < 8; ++r) crow[(size_t)r * Nc] = c[t][r];
    }
}

// ---------------------------------------------------------------------------
// Edge scatter: acc[dst] += w * h[src]. 64 threads per edge, float4 per
// thread -> coalesced 1KB row gathers; hardware fp32 global atomics for the
// accumulate (this is the memory/atomic-bound hot loop, ~800MB per layer).
// ---------------------------------------------------------------------------
__global__ __launch_bounds__(256) void scatter_edges(
    const float* __restrict__ h, const int* __restrict__ src,
    const int* __restrict__ dst, const float* __restrict__ ew,
    float* __restrict__ acc)
{
    const long long idx = (long long)blockIdx.x * 256 + threadIdx.x;
    const int e  = (int)(idx >> 6);
    const int c4 = ((int)idx & 63) << 2;
    const int s  = src[e];
    const int d  = dst[e];
    const float w = ew[e];
    const float4 hv = *(const float4*)(h + (size_t)s * DIM + c4);
    float* p = acc + (size_t)d * DIM + c4;
    unsafeAtomicAdd(p + 0, w * hv.x);
    unsafeAtomicAdd(p + 1, w * hv.y);
    unsafeAtomicAdd(p + 2, w * hv.z);
    unsafeAtomicAdd(p + 3, w * hv.w);
}

// acc = relu(acc + b)  elementwise over [N, DIM]
__global__ __launch_bounds__(256) void bias_relu(
    float* __restrict__ a, const float* __restrict__ b)
{
    const int i = blockIdx.x * 256 + threadIdx.x;
    const float v = a[i] + b[i & (DIM - 1)];
    a[i] = v > 0.0f ? v : 0.0f;
}

// row-wise softmax over k=16 logits; one thread per row
__global__ __launch_bounds__(256) void softmax16(
    const float* __restrict__ logits, float* __restrict__ out)
{
    const int r = blockIdx.x * 256 + threadIdx.x;   // grid sized exactly
    const float* row = logits + (size_t)r * KCLUST;
    float v[KCLUST];
    float mx = -3.4e38f;
    #pragma unroll
    for (int j = 0; j < KCLUST; ++j) { v[j] = row[j]; mx = v[j] > mx ? v[j] : mx; }
    float sum = 0.0f;
    #pragma unroll
    for (int j = 0; j < KCLUST; ++j) { v[j] = __expf(v[j] - mx); sum += v[j]; }
    const float inv = 1.0f / sum;
    float* orow = out + (size_t)r * KCLUST;
    #pragma unroll
    for (int j = 0; j < KCLUST; ++j) orow[j] = v[j] * inv;
}

// diag(s^T s): per-block LDS float atomics, one global atomic per column
__global__ __launch_bounds__(256) void gram_diag(
    const float* __restrict__ s, float* __restrict__ g)
{
    __shared__ float gs[KCLUST];
    if (threadIdx.x < KCLUST) gs[threadIdx.x] = 0.0f;
    __syncthreads();
    const int r = blockIdx.x * 256 + threadIdx.x;   // grid sized exactly
    const float* row = s + (size_t)r * KCLUST;
    #pragma unroll
    for (int j = 0; j < KCLUST; ++j) {
        const float v = row[j];
        atomicAdd(&gs[j], v * v);                   // ds_add_f32
    }
    __syncthreads();
    if (threadIdx.x < KCLUST) unsafeAtomicAdd(&g[threadIdx.x], gs[threadIdx.x]);
}

__global__ void finalize_loss(const float* __restrict__ g, float* __restrict__ out)
{
    if (threadIdx.x == 0) {
        float acc = 0.0f;
        #pragma unroll
        for (int j = 0; j < KCLUST; ++j) acc += sqrtf(g[j] + 1e-15f);
        out[(size_t)N_NODES * KCLUST] = -acc / sqrtf((float)N_NODES * (float)KCLUST);
    }
}

extern "C" void kernel_launch(void* const* d_in, const int* in_sizes, int n_in,
                              void* d_out, int out_size, void* d_ws, size_t ws_size,
                              hipStream_t stream)
{
    (void)in_sizes; (void)n_in; (void)out_size; (void)ws_size;

    const float* x   = (const float*)d_in[0];     // [N, d]
    const float* ew  = (const float*)d_in[1];     // [E]
    const float* W1  = (const float*)d_in[2];     // [d, d]
    const float* b1  = (const float*)d_in[3];     // [d]
    const float* W2  = (const float*)d_in[4];     // [d, d]
    const float* b2  = (const float*)d_in[5];     // [d]
    const float* Wm  = (const float*)d_in[6];     // [d, k]
    const float* bm  = (const float*)d_in[7];     // [k]
    const int*   ei  = (const int*)d_in[8];       // [2, E] row-major
    const int*   src = ei;
    const int*   dst = ei + N_EDGES;

    float* out = (float*)d_out;                   // [N*k] s, then [1] loss

    const size_t ND = (size_t)N_NODES * DIM;      // 3,145,728 floats
    float* h_gemm = (float*)d_ws;                 // GEMM output (reused)
    float* acc1   = h_gemm + ND;                  // layer-1 scatter accumulator
    float* acc2   = acc1   + ND;                  // layer-2 scatter accumulator
    float* logits = acc2   + ND;                  // [N, k]
    float* gram   = logits + (size_t)N_NODES * KCLUST;  // [k]

    // zero the accumulators (part of the captured graph -> redone each replay)
    hipMemsetAsync(acc1, 0, ND * sizeof(float), stream);
    hipMemsetAsync(acc2, 0, ND * sizeof(float), stream);
    hipMemsetAsync(gram, 0, KCLUST * sizeof(float), stream);

    const int gemm_dd_blocks = (N_NODES / 16) * (DIM / 32) / 8;     // 768 (NT=2)
    const int gemm_dk_blocks = (N_NODES / 16) * (KCLUST / 16) / 8;  // 96  (NT=1)
    const int scat_blocks    = (N_EDGES * (DIM / 4)) / 256;         // 98304
    const int elem_blocks    = (int)(ND / 256);                     // 12288
    const int row_blocks     = N_NODES / 256;                       // 48

    // layer 1: h = relu(scatter(w * (x@W1)) + b1)
    gemm_f32_wmma<2><<<gemm_dd_blocks, 256, 0, stream>>>(x, W1, nullptr, h_gemm,
                                                         N_NODES, DIM, DIM);
    scatter_edges<<<scat_blocks, 256, 0, stream>>>(h_gemm, src, dst, ew, acc1);
    bias_relu<<<elem_blocks, 256, 0, stream>>>(acc1, b1);

    // layer 2: h = relu(scatter(w * (h@W2)) + b2)
    gemm_f32_wmma<2><<<gemm_dd_blocks, 256, 0, stream>>>(acc1, W2, nullptr, h_gemm,
                                                         N_NODES, DIM, DIM);
    scatter_edges<<<scat_blocks, 256, 0, stream>>>(h_gemm, src, dst, ew, acc2);
    bias_relu<<<elem_blocks, 256, 0, stream>>>(acc2, b2);

    // cluster head: s = softmax(h@Wm + bm)
    gemm_f32_wmma<1><<<gemm_dk_blocks, 256, 0, stream>>>(acc2, Wm, bm, logits,
                                                         N_NODES, DIM, KCLUST);
    softmax16<<<row_blocks, 256, 0, stream>>>(logits, out);

    // b_loss = -sum_j sqrt(diag(s^T s)_j + EPS) / sqrt(n*k)
    gram_diag<<<row_blocks, 256, 0, stream>>>(out, gram);
    finalize_loss<<<1, 32, 0, stream>>>(gram, out);
}